// UNet4LMCD_6451040878762
// MI455X (gfx1250) — compile-verified
//
#include <hip/hip_runtime.h>
#include <hip/hip_bf16.h>

// ---------------------------------------------------------------------------
// 3D U-Net (4 levels) for MI455X / gfx1250, fp32 end-to-end.
//
// conv / tconv as implicit GEMM on V_WMMA_F32_16X16X4_F32:
//   M = 16 contiguous output-W positions, N = 16 output channels,
//   K = Cin per kernel tap; 27 taps fully unrolled at compile time.
// Interior tiles (the vast majority) take a checked-free fast path decided
// by one wave-uniform branch; border tiles take the masked path. EXEC stays
// all-ones for every WMMA in both paths (ISA requirement).
// Weights for the block's 16 output channels live in LDS as [tap][n][ci] so
// each B fragment is one aligned ds_load_b64 (pairs fuse to ds_load_2addr).
// Channel concats are free: decoder outputs are placed directly before their
// skip tensors in the workspace (NCDHW: concat along C == adjacency).
// ---------------------------------------------------------------------------

typedef float v2f __attribute__((ext_vector_type(2)));
typedef float v8f __attribute__((ext_vector_type(8)));

#ifndef TILES_PER_WAVE
#define TILES_PER_WAVE 4
#endif

// MODE: 0 = conv stride 1, 1 = conv stride 2,
//       2 = transposed conv stride 2 (lhs_dilation=2, pad (1,2), k=3 -> 2x)
// CHECK: true -> full range checks (border tiles); false -> interior tiles
//        (MODE 2 keeps only the dilation-parity mask).
template <int MODE, bool CHECK>
__device__ __forceinline__ void accum_taps(
    const float* __restrict__ in, const float* __restrict__ ldsw,
    v8f& acc, int Cin, int od, int oh, int owA, int half, int ml,
    int inD, int inH, int inW, size_t S) {
    #pragma unroll
    for (int kd = 0; kd < 3; ++kd)
    #pragma unroll
    for (int kh = 0; kh < 3; ++kh)
    #pragma unroll
    for (int kw = 0; kw < 3; ++kw) {
        const int tap = (kd * 3 + kh) * 3 + kw;
        int id, ih, iw;
        bool ok = true;
        if (MODE == 2) {
            int pd = od + kd - 1, ph = oh + kh - 1, pw = owA + kw - 1;
            ok = ((pd | ph | pw) & 1) == 0;  // dilation holes / -1 -> odd
            id = pd >> 1; ih = ph >> 1; iw = pw >> 1;
        } else {
            const int s = MODE + 1;
            id = od * s + kd - 1; ih = oh * s + kh - 1; iw = owA * s + kw - 1;
        }
        if (CHECK) {
            ok = ok && (unsigned)id < (unsigned)inD &&
                       (unsigned)ih < (unsigned)inH &&
                       (unsigned)iw < (unsigned)inW;
        }
        const bool MASKED = CHECK || (MODE == 2);
        size_t off;
        if (MASKED)
            off = ok ? (((size_t)id * inH + ih) * inW + iw) : 0;
        else
            off = ((size_t)id * inH + ih) * inW + iw;
        const float* ap = in + off + (size_t)(half * 2) * S;
        const float* bp = &ldsw[((tap << 4) + ml) * Cin + half * 2];

        if (MASKED) {
            #pragma unroll 2
            for (int c0 = 0; c0 < Cin; c0 += 4) {
                float a0 = ap[(size_t)c0 * S];
                float a1 = ap[(size_t)c0 * S + S];
                v2f a, b;
                a[0] = ok ? a0 : 0.0f;
                a[1] = ok ? a1 : 0.0f;
                b = *(const v2f*)(bp + c0);
                acc = __builtin_amdgcn_wmma_f32_16x16x4_f32(
                    false, a, false, b, (short)0, acc, false, false);
            }
        } else {
            #pragma unroll 2
            for (int c0 = 0; c0 < Cin; c0 += 4) {
                v2f a, b;
                a[0] = ap[(size_t)c0 * S];
                a[1] = ap[(size_t)c0 * S + S];
                b = *(const v2f*)(bp + c0);
                acc = __builtin_amdgcn_wmma_f32_16x16x4_f32(
                    false, a, false, b, (short)0, acc, false, false);
            }
        }
    }
}

template <int MODE>
__global__ __launch_bounds__(256)
void conv3d_wmma(const float* __restrict__ in, const float* __restrict__ wgt,
                 float* __restrict__ out, int Cin, int Cout,
                 int inD, int inH, int inW, int outD, int outH, int outW) {
    extern __shared__ float lds[];  // [(tap*16 + n) * Cin + ci]
    const int tid = threadIdx.x;
    const int coT = blockIdx.y;

    // ---- stage this block's weight tile (16 out-channels) into LDS ----
    const int wTotal = 27 * 16 * Cin;
    for (int idx = tid; idx < wTotal; idx += 256) {
        int n   = idx & 15;
        int t2  = idx >> 4;
        int tap = t2 % 27;
        int ci  = t2 / 27;
        int co  = (coT << 4) + n;
        float v = (co < Cout) ? wgt[((size_t)co * Cin + ci) * 27 + tap] : 0.0f;
        lds[((tap << 4) + n) * Cin + ci] = v;
    }
    __syncthreads();

    const int lane = tid & 31;
    const int wave = tid >> 5;
    const int half = lane >> 4;  // 0 -> K={0,1}, 1 -> K={2,3} within chunk
    const int ml   = lane & 15;  // A row (M) / B col (N)

    const int    wTiles = outW >> 4;
    const int    sTiles = outD * outH * wTiles;  // <= 131072: 32-bit is fine
    const size_t S      = (size_t)inD * inH * inW;
    const int    step   = (int)gridDim.x * 8;
    const int    co     = (coT << 4) + ml;

    for (int st = (int)blockIdx.x * 8 + wave; st < sTiles; st += step) {
        int wT = st % wTiles;
        int r  = st / wTiles;
        int oh = r % outH;
        int od = r / outH;
        const int owBase = wT << 4;
        const int owA    = owBase + ml;

        // Wave-uniform interior test (no clipping possible anywhere in tile).
        bool interior;
        if (MODE == 1)
            interior = (od >= 1) && (oh >= 1) && (wT >= 1);
        else
            interior = (od >= 1) && (od <= outD - 2) &&
                       (oh >= 1) && (oh <= outH - 2) &&
                       (wT >= 1) && (wT <= wTiles - 2);

        v8f acc = {};
        if (interior)
            accum_taps<MODE, false>(in, lds, acc, Cin, od, oh, owA, half, ml,
                                    inD, inH, inW, S);
        else
            accum_taps<MODE, true>(in, lds, acc, Cin, od, oh, owA, half, ml,
                                   inD, inH, inW, S);

        if (co < Cout) {
            float* orow = out + (((size_t)co * outD + od) * outH + oh) * outW
                        + owBase + half * 8;
            *(float4*)(orow)     = make_float4(acc[0], acc[1], acc[2], acc[3]);
            *(float4*)(orow + 4) = make_float4(acc[4], acc[5], acc[6], acc[7]);
        }
    }
}

// Per-channel mean/var over S spatial elements (N=1). S is a multiple of 4.
__global__ __launch_bounds__(256)
void bn_stats(const float* __restrict__ x, float* __restrict__ mean,
              float* __restrict__ var, int S) {
    const int c = blockIdx.x;
    const float4* p = (const float4*)(x + (size_t)c * S);
    const int S4 = S >> 2;
    float s = 0.0f, s2 = 0.0f;
    for (int i = threadIdx.x; i < S4; i += 256) {
        float4 v = p[i];
        s  += v.x + v.y + v.z + v.w;
        s2 += v.x * v.x + v.y * v.y + v.z * v.z + v.w * v.w;
    }
    __shared__ float sh[256], sh2[256];
    sh[threadIdx.x] = s; sh2[threadIdx.x] = s2;
    __syncthreads();
    for (int off = 128; off > 0; off >>= 1) {
        if (threadIdx.x < off) {
            sh[threadIdx.x]  += sh[threadIdx.x + off];
            sh2[threadIdx.x] += sh2[threadIdx.x + off];
        }
        __syncthreads();
    }
    if (threadIdx.x == 0) {
        float inv = 1.0f / (float)S;
        float m = sh[0] * inv;
        mean[c] = m;
        var[c]  = sh2[0] * inv - m * m;
    }
}

// In-place y = relu((x - mu) * rsqrt(var + eps) * g + b)
__global__ __launch_bounds__(256)
void bn_apply(float* __restrict__ x, const float* __restrict__ mean,
              const float* __restrict__ var, const float* __restrict__ g,
              const float* __restrict__ b, int S, long total) {
    long i = (long)blockIdx.x * 256 + threadIdx.x;
    if (i >= total) return;
    int c = (int)(i / S);
    float xh = (x[i] - mean[c]) * rsqrtf(var[c] + 1e-5f);
    float y  = xh * g[c] + b[c];
    x[i] = fmaxf(y, 0.0f);
}

extern "C" void kernel_launch(void* const* d_in, const int* in_sizes, int n_in,
                              void* d_out, int out_size, void* d_ws, size_t ws_size,
                              hipStream_t stream) {
    (void)in_sizes; (void)n_in; (void)out_size; (void)ws_size;

    const float* x    = (const float*)d_in[0];
    const float* w0   = (const float*)d_in[1];
    const float* g0   = (const float*)d_in[2];
    const float* b0   = (const float*)d_in[3];
    const float* w1   = (const float*)d_in[4];
    const float* g1   = (const float*)d_in[5];
    const float* b1   = (const float*)d_in[6];
    const float* w2   = (const float*)d_in[7];
    const float* g2   = (const float*)d_in[8];
    const float* b2   = (const float*)d_in[9];
    const float* w3   = (const float*)d_in[10];
    const float* g3   = (const float*)d_in[11];
    const float* b3   = (const float*)d_in[12];
    const float* wt3  = (const float*)d_in[13];
    const float* gt3  = (const float*)d_in[14];
    const float* bt3  = (const float*)d_in[15];
    const float* wt2  = (const float*)d_in[16];
    const float* gt2  = (const float*)d_in[17];
    const float* bt2  = (const float*)d_in[18];
    const float* wt1  = (const float*)d_in[19];
    const float* gt1  = (const float*)d_in[20];
    const float* bt1  = (const float*)d_in[21];
    const float* wout = (const float*)d_in[22];
    float* outp = (float*)d_out;

    const size_t S128 = 128ull * 128 * 128;
    const size_t S64  = 64ull * 64 * 64;
    const size_t S32  = 32ull * 32 * 32;
    const size_t S16  = 16ull * 16 * 16;

    // Workspace (floats); decoder outputs directly precede their skips:
    //   region C: [ d1 : 24ch @128^3 ][ s1 : 8ch @128^3 ]  -> 32ch tensor
    //   region B: [ d2 : 32ch @ 64^3 ][ s2 :16ch @ 64^3 ]  -> 48ch tensor
    //   region A: [ d3 : 32ch @ 32^3 ][ s4 :32ch @ 32^3 ]  -> 64ch tensor
    float* wsf = (float*)d_ws;
    float* d1  = wsf;
    float* s1  = d1 + 24 * S128;
    float* d2  = s1 + 8 * S128;
    float* s2  = d2 + 32 * S64;
    float* d3  = s2 + 16 * S64;
    float* s4  = d3 + 32 * S32;
    float* s8  = s4 + 32 * S32;
    float* mu  = s8 + 64 * S16;
    float* va  = mu + 64;

    auto conv = [&](const float* ip, const float* wp, float* op,
                    int Cin, int Cout, int iD, int iH, int iW,
                    int oD, int oH, int oW, int mode) {
        long   sTiles  = (long)oD * oH * (oW >> 4);
        int    coTiles = (Cout + 15) >> 4;
        long   blocks  = (sTiles + 8 * TILES_PER_WAVE - 1) / (8 * TILES_PER_WAVE);
        size_t ldsB    = (size_t)27 * 16 * Cin * sizeof(float);
        dim3 grid((unsigned)blocks, (unsigned)coTiles);
        if (mode == 0)
            conv3d_wmma<0><<<grid, dim3(256), ldsB, stream>>>(
                ip, wp, op, Cin, Cout, iD, iH, iW, oD, oH, oW);
        else if (mode == 1)
            conv3d_wmma<1><<<grid, dim3(256), ldsB, stream>>>(
                ip, wp, op, Cin, Cout, iD, iH, iW, oD, oH, oW);
        else
            conv3d_wmma<2><<<grid, dim3(256), ldsB, stream>>>(
                ip, wp, op, Cin, Cout, iD, iH, iW, oD, oH, oW);
    };
    auto bn = [&](float* t, const float* g, const float* b, int C, size_t S) {
        bn_stats<<<dim3((unsigned)C), dim3(256), 0, stream>>>(t, mu, va, (int)S);
        long total  = (long)C * (long)S;
        long blocks = (total + 255) / 256;
        bn_apply<<<dim3((unsigned)blocks), dim3(256), 0, stream>>>(
            t, mu, va, g, b, (int)S, total);
    };

    // ---- encoder ----
    conv(x,  w0, s1,  8,  8, 128, 128, 128, 128, 128, 128, 0);
    bn(s1, g0, b0, 8, S128);
    conv(s1, w1, s2,  8, 16, 128, 128, 128,  64,  64,  64, 1);
    bn(s2, g1, b1, 16, S64);
    conv(s2, w2, s4, 16, 32,  64,  64,  64,  32,  32,  32, 1);
    bn(s4, g2, b2, 32, S32);
    conv(s4, w3, s8, 32, 64,  32,  32,  32,  16,  16,  16, 1);
    bn(s8, g3, b3, 64, S16);

    // ---- decoder (concats free via adjacency) ----
    conv(s8, wt3, d3, 64, 32, 16, 16, 16, 32, 32, 32, 2);
    bn(d3, gt3, bt3, 32, S32);
    conv(d3, wt2, d2, 64, 32, 32, 32, 32, 64, 64, 64, 2);    // reads [d3|s4]
    bn(d2, gt2, bt2, 32, S64);
    conv(d2, wt1, d1, 48, 24, 64, 64, 64, 128, 128, 128, 2); // reads [d2|s2]
    bn(d1, gt1, bt1, 24, S128);

    // ---- output head: reads [d1|s1] (32ch @128^3) ----
    conv(d1, wout, outp, 32, 2, 128, 128, 128, 128, 128, 128, 0);
}